// IntraAgg_66520453480950
// MI455X (gfx1250) — compile-verified
//
#include <hip/hip_runtime.h>

// IntraAgg on MI455X (gfx1250):
//   out[r] = [ mean_k emb[idx[r,k]] , self[r] - mean_k emb[idx[r,k]] ]
// Memory-bound: ~819 MB of gathered row reads (L2-resident: table = 102.4 MB
// < 192 MB L2), ~180 MB HBM floor (~8 us @ 23.3 TB/s). No shared-operand
// matmul structure -> WMMA inapplicable (block-diagonal mask matmul, no byte
// savings). Use the CDNA5 Tensor Data Mover in GATHER mode (8 x 512B rows per
// descriptor -> LDS, TENSORcnt-tracked), double-buffered per wave, with NT
// hints on streaming traffic so the embedding table stays hot in L2.

typedef __attribute__((ext_vector_type(4))) unsigned int v4u;
typedef __attribute__((ext_vector_type(8))) int          v8i;
typedef __attribute__((ext_vector_type(4))) int          v4i;
typedef __attribute__((ext_vector_type(4))) float        v4f;

#define DIMS 128              // embedding dim (f32)
#define KN   32               // neighbors per row
#define WPB  2                // waves per block
#define SLAB 16384u           // 32 rows * 512 B gathered per output row

// One TDM gather descriptor: 8 rows of the embedding table into the 4 KB of LDS starting at lds_off.
// D# layout per CDNA5 ISA (gather mode, 32-bit indices).
// clang-23 builtin arity is 6: (v4u g0, v8i g1, v4i, v4i, v8i, i32 cpol).
// Row indices are supplied both as the two quads and as the combined octet so
// either operand-group interpretation receives them.
__device__ __forceinline__ void tdm_gather8(unsigned lds_off,
                                            unsigned long long gaddr,
                                            const int* __restrict__ ip,
                                            unsigned nembed) {
  v4u g0;
  g0.x = 0xC0000001u;                                   // count=1, gather_mode=1, 32b idx
  g0.y = lds_off;                                       // LDS byte address
  g0.z = (unsigned)(gaddr & 0xFFFFFFFFull);             // global_addr[31:0]
  g0.w = (unsigned)((gaddr >> 32) & 0x1FFFFFFull)       // global_addr[56:32]
         | (2u << 30);                                  // type=2 ("image")
  v8i g1;
  g1[0] = 0x00020000;                                   // wg_mask=0, data_size=2 (4B)
  g1[1] = (int)(((unsigned)DIMS & 0xFFFFu) << 16);      // tensor_dim0[15:0] @ bits 63:48
  g1[2] = (int)((nembed & 0xFFFFu) << 16);              // tensor_dim1[15:0] @ bits 95:80
  g1[3] = (int)((((unsigned)DIMS) << 16)                // tile_dim0=128   @ bits 127:112
                | ((nembed >> 16) & 0xFFFFu));          // tensor_dim1[31:16]
  g1[4] = 8;                                            // tile_dim1 = 8 valid indices
  g1[5] = DIMS;                                         // tensor_dim0_stride = 128 elems
  g1[6] = 0;
  g1[7] = 0;
  v4i g2; g2[0] = ip[0]; g2[1] = ip[1]; g2[2] = ip[2]; g2[3] = ip[3];
  v4i g3; g3[0] = ip[4]; g3[1] = ip[5]; g3[2] = ip[6]; g3[3] = ip[7];
  v8i g23;
  g23[0] = ip[0]; g23[1] = ip[1]; g23[2] = ip[2]; g23[3] = ip[3];
  g23[4] = ip[4]; g23[5] = ip[5]; g23[6] = ip[6]; g23[7] = ip[7];
  __builtin_amdgcn_tensor_load_to_lds(g0, g1, g2, g3, g23, /*cpol=*/0);
}

// Issue all 32 neighbor rows of output row r into a 16KB LDS slab (4 TDM ops).
__device__ __forceinline__ void issue_row(unsigned lds_off,
                                          const int* __restrict__ ip,
                                          unsigned long long gaddr,
                                          unsigned nembed) {
  tdm_gather8(lds_off          , gaddr, ip     , nembed);
  tdm_gather8(lds_off +  4096u , gaddr, ip + 8 , nembed);
  tdm_gather8(lds_off +  8192u , gaddr, ip + 16, nembed);
  tdm_gather8(lds_off + 12288u , gaddr, ip + 24, nembed);
}

__global__ void __launch_bounds__(WPB * 32, 1)
IntraAgg_66520453480950_kernel(const float* __restrict__ emb,
                               const float* __restrict__ selff,
                               const int*   __restrict__ nidx,
                               float*       __restrict__ out,
                               unsigned Brows, unsigned Nembed,
                               unsigned totalWaves) {
  extern __shared__ char smem[];
  const unsigned lane = threadIdx.x & 31u;
  // Force wave-uniformity so neighbor-index loads become scalar (SMEM) loads.
  const unsigned wave = __builtin_amdgcn_readfirstlane(threadIdx.x >> 5);
  const unsigned gw   = __builtin_amdgcn_readfirstlane(blockIdx.x * WPB + wave);
  const unsigned lds_base = (unsigned)(unsigned long long)(uintptr_t)smem
                            + wave * (2u * SLAB);
  const unsigned long long embA = (unsigned long long)(uintptr_t)emb;

  unsigned r = gw;
  if (r >= Brows) return;

  unsigned buf = 0;
  issue_row(lds_base, nidx + (unsigned long long)r * KN, embA, Nembed);

  for (;;) {
    const unsigned rn = r + totalWaves;
    const bool more = rn < Brows;
    if (more) {
      issue_row(lds_base + (buf ^ 1u) * SLAB,
                nidx + (unsigned long long)rn * KN, embA, Nembed);
      __builtin_amdgcn_s_wait_tensorcnt(4);   // older buffer's 4 ops done (in-order)
    } else {
      __builtin_amdgcn_s_wait_tensorcnt(0);
    }

    // Reduce 32 gathered rows from LDS; lane t owns float columns 4t..4t+3.
    const v4f* rp = (const v4f*)(smem + wave * (2u * SLAB) + buf * SLAB);
    v4f acc = rp[lane];
#pragma unroll
    for (int k = 1; k < KN; ++k)
      acc += rp[(unsigned)k * 32u + lane];    // 512B stride per row: conflict-free
    const v4f m = acc * (1.0f / (float)KN);

    const v4f* sp = (const v4f*)(selff + (unsigned long long)r * DIMS);
    const v4f  s  = __builtin_nontemporal_load(sp + lane);

    v4f* op = (v4f*)(out + (unsigned long long)r * (2u * DIMS));
    __builtin_nontemporal_store(m,     op + lane);        // feats_1
    __builtin_nontemporal_store(s - m, op + lane + 32u);  // feats_2

    if (!more) break;
    r = rn;
    buf ^= 1u;
  }
}

extern "C" void kernel_launch(void* const* d_in, const int* in_sizes, int n_in,
                              void* d_out, int out_size, void* d_ws, size_t ws_size,
                              hipStream_t stream) {
  const float* emb   = (const float*)d_in[0];
  const float* selff = (const float*)d_in[1];
  const int*   nidx  = (const int*)d_in[2];
  float*       out   = (float*)d_out;

  const unsigned Nembed = (unsigned)(in_sizes[0] / DIMS);
  const unsigned Brows  = (unsigned)(in_sizes[1] / DIMS);

  const unsigned blocks = 2048;                 // 4096 waves -> ~12 rows/wave
  const unsigned totalWaves = blocks * WPB;
  const unsigned shmem = WPB * 2u * SLAB;       // 64 KB: 2 waves x double buffer

  IntraAgg_66520453480950_kernel<<<blocks, WPB * 32, shmem, stream>>>(
      emb, selff, nidx, out, Brows, Nembed, totalWaves);
}